// SAGEEncoder_65171833749590
// MI455X (gfx1250) — compile-verified
//
#include <hip/hip_runtime.h>
#include <hip/hip_bf16.h>

#define N_NODES 100000
#define N_EDGES 600000
#define D 128

typedef __attribute__((ext_vector_type(2))) float v2f;
typedef __attribute__((ext_vector_type(8))) float v8f;

// ---------------- utility: zero a float buffer ----------------
__global__ void zero_f32(float* __restrict__ p, int n) {
  int i = blockIdx.x * blockDim.x + threadIdx.x;
  int stride = gridDim.x * blockDim.x;
  for (; i < n; i += stride) p[i] = 0.0f;
}

// ---------------- degree count (edges fixed across layers) ----------------
__global__ void deg_kernel(const int* __restrict__ ei, float* __restrict__ deg) {
  int e = blockIdx.x * blockDim.x + threadIdx.x;
  if (e < N_EDGES) {
    int dst = ei[N_EDGES + e];
    unsafeAtomicAdd(&deg[dst], 1.0f);
  }
}

__global__ void recip_kernel(float* __restrict__ deg) {
  int i = blockIdx.x * blockDim.x + threadIdx.x;
  if (i < N_NODES) {
    float d = deg[i];
    deg[i] = 1.0f / fmaxf(d, 1.0f);
  }
}

// ---------------- edge scatter: one wave32 per edge, float4 per lane ----------------
__global__ void scatter_kernel(const int* __restrict__ ei, const float* __restrict__ X,
                               float* __restrict__ S) {
  int gwave = (blockIdx.x * blockDim.x + threadIdx.x) >> 5;
  int lane = threadIdx.x & 31;
  if (gwave >= N_EDGES) return;
  int src = ei[gwave];
  int dst = ei[N_EDGES + gwave];
  const float4 v = *(const float4*)(X + (size_t)src * D + lane * 4);
  float* o = S + (size_t)dst * D + lane * 4;
  unsafeAtomicAdd(o + 0, v.x);
  unsafeAtomicAdd(o + 1, v.y);
  unsafeAtomicAdd(o + 2, v.z);
  unsafeAtomicAdd(o + 3, v.w);
}

// ---------------- fused SAGE layer GEMM via V_WMMA_F32_16X16X4_F32 ----------------
// out[n,:] = opt_relu( (S[n,:]*rdeg[n]) @ Wl + bl + H[n,:] @ Wr )
// Block = 256 threads (8 waves). Each block: 16 nodes x 128 output cols.
// Each wave computes one 16x16 tile: 32 + 32 chained f32 WMMAs.
// H and out may alias (in-place layer 3): each block reads only its own 16 rows
// of H (staged to LDS before the barrier) and writes only those same rows.
template <bool RELU>
__global__ __launch_bounds__(256) void sage_gemm(
    const float* __restrict__ S, const float* __restrict__ rdeg,
    const float* H, const float* __restrict__ Wl,
    const float* __restrict__ bl, const float* __restrict__ Wr,
    float* out) {
  __shared__ float mt[16][132];  // mean tile, padded row stride vs 64 banks
  __shared__ float ht[16][132];  // self tile

  const int tid = threadIdx.x;
  const int rowbase = blockIdx.x * 16;

  // cooperative stage: 16x128 per tile, 2048 elems, 8 per thread, coalesced
  for (int i = tid; i < 16 * D; i += 256) {
    int r = i >> 7, c = i & 127;
    mt[r][c] = S[(size_t)(rowbase + r) * D + c] * rdeg[rowbase + r];
    ht[r][c] = H[(size_t)(rowbase + r) * D + c];
  }
  __syncthreads();

  const int lane = tid & 31;
  const int wavecol = (tid >> 5) << 4;  // 16-col tile per wave
  const int nm = lane & 15;             // M index for A, N index for B/C/D
  const int khalf = (lane >> 4) << 1;   // lanes 16-31 hold K+2,K+3

  // accumulator = bias broadcast down the M dimension
  v8f acc;
  {
    float b0 = bl[wavecol + nm];
#pragma unroll
    for (int j = 0; j < 8; ++j) acc[j] = b0;
  }

  // mean @ Wl
#pragma unroll 8
  for (int k = 0; k < D; k += 4) {
    int kk = k + khalf;
    v2f a, b;
    a.x = mt[nm][kk];
    a.y = mt[nm][kk + 1];
    b.x = Wl[kk * D + wavecol + nm];
    b.y = Wl[(kk + 1) * D + wavecol + nm];
    acc = __builtin_amdgcn_wmma_f32_16x16x4_f32(false, a, false, b, (short)0, acc,
                                                false, false);
  }
  // self @ Wr
#pragma unroll 8
  for (int k = 0; k < D; k += 4) {
    int kk = k + khalf;
    v2f a, b;
    a.x = ht[nm][kk];
    a.y = ht[nm][kk + 1];
    b.x = Wr[kk * D + wavecol + nm];
    b.y = Wr[(kk + 1) * D + wavecol + nm];
    acc = __builtin_amdgcn_wmma_f32_16x16x4_f32(false, a, false, b, (short)0, acc,
                                                false, false);
  }

  // C/D layout: VGPR j -> M=j (lanes 0-15) or M=j+8 (lanes 16-31), N=lane&15
  const int moff = (lane >> 4) << 3;
#pragma unroll
  for (int j = 0; j < 8; ++j) {
    float v = acc[j];
    if (RELU) v = fmaxf(v, 0.0f);
    out[(size_t)(rowbase + j + moff) * D + wavecol + nm] = v;
  }
}

extern "C" void kernel_launch(void* const* d_in, const int* in_sizes, int n_in,
                              void* d_out, int out_size, void* d_ws, size_t ws_size,
                              hipStream_t stream) {
  (void)in_sizes; (void)n_in; (void)out_size; (void)ws_size;

  const float* x   = (const float*)d_in[0];
  const int*   ei  = (const int*)d_in[1];   // [2, E]: src row then dst row
  // d_in[2] = edge_attr (unused by reference layers)
  const float* Wl1 = (const float*)d_in[3];
  const float* bl1 = (const float*)d_in[4];
  const float* Wr1 = (const float*)d_in[5];
  const float* Wl2 = (const float*)d_in[6];
  const float* bl2 = (const float*)d_in[7];
  const float* Wr2 = (const float*)d_in[8];
  const float* Wl3 = (const float*)d_in[9];
  const float* bl3 = (const float*)d_in[10];
  const float* Wr3 = (const float*)d_in[11];
  float* out = (float*)d_out;

  // workspace layout: S [N*D] | h1 [N*D] | rdeg [N]
  float* S    = (float*)d_ws;
  float* h1   = S + (size_t)N_NODES * D;
  float* rdeg = h1 + (size_t)N_NODES * D;

  const int ND = N_NODES * D;
  const int edge_blocks = N_EDGES / 8;     // 8 edges (waves) per 256-thread block
  const int gemm_blocks = N_NODES / 16;    // 100000 / 16 = 6250 exact

  // degrees (structure-only, once)
  zero_f32<<<128, 256, 0, stream>>>(rdeg, N_NODES);
  deg_kernel<<<(N_EDGES + 255) / 256, 256, 0, stream>>>(ei, rdeg);
  recip_kernel<<<(N_NODES + 255) / 256, 256, 0, stream>>>(rdeg);

  // layer 1: x -> h1 (relu)
  zero_f32<<<2048, 256, 0, stream>>>(S, ND);
  scatter_kernel<<<edge_blocks, 256, 0, stream>>>(ei, x, S);
  sage_gemm<true><<<gemm_blocks, 256, 0, stream>>>(S, rdeg, x, Wl1, bl1, Wr1, h1);

  // layer 2: h1 -> out (relu); out buffer doubles as h2 storage
  zero_f32<<<2048, 256, 0, stream>>>(S, ND);
  scatter_kernel<<<edge_blocks, 256, 0, stream>>>(ei, h1, S);
  sage_gemm<true><<<gemm_blocks, 256, 0, stream>>>(S, rdeg, h1, Wl2, bl2, Wr2, out);

  // layer 3: out -> out in-place (no relu); per-tile read-before-write is safe
  zero_f32<<<2048, 256, 0, stream>>>(S, ND);
  scatter_kernel<<<edge_blocks, 256, 0, stream>>>(ei, out, S);
  sage_gemm<false><<<gemm_blocks, 256, 0, stream>>>(S, rdeg, out, Wl3, bl3, Wr3, out);
}